// ChebC_9783935500637
// MI455X (gfx1250) — compile-verified
//
#include <hip/hip_runtime.h>
#include <hip/hip_bf16.h>

// ---------------------------------------------------------------------------
// ChebConv GNN on MI455X (gfx1250, wave32).
// Per-graph dense reformulation: each 64-node graph's normalized Laplacian is
// materialized as a dense 64x64 matrix in LDS; all propagation and weight
// multiplies become dense GEMMs executed with V_WMMA_F32_16X16X4_F32 (exact
// fp32 matrix pipe -- compute is nowhere near the WMMA ceiling, so we keep
// full precision). Tile loops are compile-time unrolled so no EXEC-masked
// control flow surrounds the WMMA regions (EXEC all-ones requirement).
// ---------------------------------------------------------------------------

#define G_   512
#define NPG_ 64
#define N_   (G_ * NPG_)
#define EPG_ 2048          // NPG * DEG
#define F3_  64

typedef float v2f __attribute__((ext_vector_type(2)));
typedef float v8f __attribute__((ext_vector_type(8)));

// ---- WMMA f32 16x16x4 fragment helpers ------------------------------------
// A fragment: 16x4 tile, rows m0..m0+15, cols k0..k0+3 of row-major src (ld).
// Layout (ISA 7.12.2 "32-bit A-Matrix 16x4"): lane<16 -> VGPR0=K0,VGPR1=K1;
// lane>=16 -> VGPR0=K2,VGPR1=K3; M = lane&15.
__device__ __forceinline__ v2f frag_a(const float* __restrict__ s, int ld,
                                      int m0, int k0, int lane) {
    int m  = m0 + (lane & 15);
    int kb = k0 + ((lane >> 4) << 1);
    return *(const v2f*)(s + m * ld + kb);   // kb even, 8B aligned
}
// B fragment: 4x16 tile, rows k0..k0+3, cols n0..n0+15 of row-major src (ld).
// Mirrors A: lane&15 = N; half-wave selects K pair (VGPR0=K2h, VGPR1=K2h+1).
__device__ __forceinline__ v2f frag_b(const float* __restrict__ s, int ld,
                                      int k0, int n0, int lane) {
    int n  = n0 + (lane & 15);
    int kb = k0 + ((lane >> 4) << 1);
    v2f b;
    b.x = s[kb * ld + n];
    b.y = s[(kb + 1) * ld + n];
    return b;
}
__device__ __forceinline__ v8f wmma4(v2f a, v2f b, v8f c) {
    return __builtin_amdgcn_wmma_f32_16x16x4_f32(
        /*neg_a=*/false, a, /*neg_b=*/false, b,
        /*c_mod=*/(short)0, c, /*reuse_a=*/false, /*reuse_b=*/false);
}

// ---------------------------------------------------------------------------
// Kernel 1: build dense A_hat[g] (64x64) per graph.
//   A_hat[c][r] = sum over edges (r,c) of -(dinv[r]*w*dinv[c])
// Edges for graph g are contiguous: [g*2048, (g+1)*2048); node ids & 63 give
// local indices. All accumulation in LDS (ds_add_f32), no global atomics.
// ---------------------------------------------------------------------------
__global__ __launch_bounds__(256) void build_ahat(
    const int* __restrict__ row, const int* __restrict__ col,
    const float* __restrict__ ew, float* __restrict__ Ahat) {
    __shared__ float sdeg[NPG_];
    __shared__ float sdinv[NPG_];
    __shared__ float sA[NPG_ * NPG_];

    const int g = blockIdx.x;
    const int tid = threadIdx.x;
    const int e0 = g * EPG_;

    for (int i = tid; i < NPG_ * NPG_; i += 256) sA[i] = 0.0f;
    if (tid < NPG_) sdeg[tid] = 0.0f;
    __syncthreads();

    for (int e = tid; e < EPG_; e += 256)
        atomicAdd(&sdeg[row[e0 + e] & 63], ew[e0 + e]);
    __syncthreads();

    if (tid < NPG_) {
        float d = sdeg[tid];
        sdinv[tid] = (d > 0.0f) ? rsqrtf(d) : 0.0f;
    }
    __syncthreads();

    for (int e = tid; e < EPG_; e += 256) {
        int r = row[e0 + e] & 63;
        int c = col[e0 + e] & 63;
        float v = -(sdinv[r] * ew[e0 + e] * sdinv[c]);
        atomicAdd(&sA[c * NPG_ + r], v);
    }
    __syncthreads();

    float* dst = Ahat + g * (NPG_ * NPG_);
    for (int i = tid; i < NPG_ * NPG_; i += 256) dst[i] = sA[i];
}

// ---------------------------------------------------------------------------
// Dense 64x64 @ 64xFN matmul in LDS via WMMA. Optionally D = 2*A*B - Sub.
// 8 waves split the (4 x FN/16) output tiles. NT/8 is a compile-time
// constant, so the tile loop fully unrolls: no divergent control flow, EXEC
// stays all-ones through every WMMA.
// ---------------------------------------------------------------------------
template <int FN, bool SCALE_SUB>
__device__ __forceinline__ void prop_mm(
    const float* __restrict__ A, const float* __restrict__ B,
    const float* __restrict__ Sub, float* __restrict__ Out,
    int wid, int lane) {
    constexpr int NTN = FN / 16;
    constexpr int NT  = 4 * NTN;           // 16 (FN=64) or 32 (FN=128)
    static_assert(NT % 8 == 0, "tiles must split evenly over 8 waves");
#pragma unroll
    for (int i = 0; i < NT / 8; ++i) {
        const int t  = wid + i * 8;
        const int m0 = (t / NTN) * 16;
        const int n0 = (t % NTN) * 16;
        v8f acc = {};
#pragma unroll
        for (int k0 = 0; k0 < 64; k0 += 4) {
            v2f a = frag_a(A, 64, m0, k0, lane);
            v2f b = frag_b(B, FN, k0, n0, lane);
            acc = wmma4(a, b, acc);
        }
        const int coln = n0 + (lane & 15);
        const int rb   = m0 + ((lane >> 4) ? 8 : 0);
#pragma unroll
        for (int j = 0; j < 8; ++j) {
            float v = acc[j];
            if (SCALE_SUB) v = 2.0f * v - Sub[(rb + j) * FN + coln];
            Out[(rb + j) * FN + coln] = v;
        }
    }
}

// ---------------------------------------------------------------------------
// Kernel 2: one ChebConv layer for one graph per workgroup.
//   Tx1 = A*Tx0 ; Tx2 = 2*A*Tx1 - Tx0 ; Y = Tx0*W0 + Tx1*W1 + Tx2*W2 + b
// ---------------------------------------------------------------------------
template <int FIN, int FOUT, bool RELU>
__global__ __launch_bounds__(256) void cheb_layer(
    const float* __restrict__ X, const float* __restrict__ Ahat,
    const float* __restrict__ W, const float* __restrict__ bias,
    float* __restrict__ Y) {
    __shared__ float sA[NPG_ * NPG_];
    __shared__ float sT0[NPG_ * FIN];
    __shared__ float sT1[NPG_ * FIN];
    __shared__ float sT2[NPG_ * FIN];

    const int g    = blockIdx.x;
    const int tid  = threadIdx.x;
    const int lane = tid & 31;
    const int wid  = tid >> 5;

    const float* __restrict__ Ag = Ahat + g * (NPG_ * NPG_);
    const float* __restrict__ Xg = X + g * (NPG_ * FIN);
    for (int i = tid; i < NPG_ * NPG_; i += 256) sA[i] = Ag[i];
    for (int i = tid; i < NPG_ * FIN; i += 256) sT0[i] = Xg[i];
    __syncthreads();

    prop_mm<FIN, false>(sA, sT0, nullptr, sT1, wid, lane);
    __syncthreads();
    prop_mm<FIN, true>(sA, sT1, sT0, sT2, wid, lane);
    __syncthreads();

    // Output GEMM: 64xFOUT = sum_k (64xFIN)(FINxFOUT); W streamed from L2.
    const float* __restrict__ W0 = W;
    const float* __restrict__ W1 = W + FIN * FOUT;
    const float* __restrict__ W2 = W + 2 * FIN * FOUT;
    float* __restrict__ Yg = Y + g * (NPG_ * FOUT);

    constexpr int NTN = FOUT / 16;
    constexpr int NT  = 4 * NTN;
    static_assert(NT % 8 == 0, "tiles must split evenly over 8 waves");
#pragma unroll
    for (int i = 0; i < NT / 8; ++i) {
        const int t  = wid + i * 8;
        const int m0 = (t / NTN) * 16;
        const int n0 = (t % NTN) * 16;
        v8f acc = {};
#pragma unroll 4
        for (int k0 = 0; k0 < FIN; k0 += 4) {
            acc = wmma4(frag_a(sT0, FIN, m0, k0, lane),
                        frag_b(W0, FOUT, k0, n0, lane), acc);
            acc = wmma4(frag_a(sT1, FIN, m0, k0, lane),
                        frag_b(W1, FOUT, k0, n0, lane), acc);
            acc = wmma4(frag_a(sT2, FIN, m0, k0, lane),
                        frag_b(W2, FOUT, k0, n0, lane), acc);
        }
        const int coln = n0 + (lane & 15);
        const int rb   = m0 + ((lane >> 4) ? 8 : 0);
        const float bb = bias[coln];
#pragma unroll
        for (int j = 0; j < 8; ++j) {
            float v = acc[j] + bb;
            if (RELU) v = fmaxf(v, 0.0f);
            Yg[(rb + j) * FOUT + coln] = v;
        }
    }
}

// ---------------------------------------------------------------------------
// Kernel 3: mean/max readout over each graph's 64 nodes + FC (128 -> 64).
// ---------------------------------------------------------------------------
__global__ __launch_bounds__(64) void readout(
    const float* __restrict__ H, const float* __restrict__ fcw,
    const float* __restrict__ fcb, float* __restrict__ out) {
    __shared__ float pooled[2 * F3_];
    const int g = blockIdx.x;
    const int j = threadIdx.x;       // 0..63, one output feature each
    const float* __restrict__ hg = H + g * (NPG_ * F3_);

    float s = 0.0f, mx = -3.402823466e38f;
    for (int i = 0; i < NPG_; ++i) {
        float v = hg[i * F3_ + j];
        s += v;
        mx = fmaxf(mx, v);
    }
    pooled[j]       = s * (1.0f / (float)NPG_);
    pooled[F3_ + j] = mx;
    __syncthreads();

    float acc = fcb[j];
#pragma unroll 8
    for (int i = 0; i < 2 * F3_; ++i) acc = fmaf(pooled[i], fcw[i * F3_ + j], acc);
    out[g * F3_ + j] = acc;
}

// ---------------------------------------------------------------------------
extern "C" void kernel_launch(void* const* d_in, const int* in_sizes, int n_in,
                              void* d_out, int out_size, void* d_ws, size_t ws_size,
                              hipStream_t stream) {
    (void)in_sizes; (void)n_in; (void)out_size; (void)ws_size;
    const float* x   = (const float*)d_in[0];
    const float* ew  = (const float*)d_in[1];
    const float* W1  = (const float*)d_in[2];
    const float* b1  = (const float*)d_in[3];
    const float* W2  = (const float*)d_in[4];
    const float* b2  = (const float*)d_in[5];
    const float* W3  = (const float*)d_in[6];
    const float* b3  = (const float*)d_in[7];
    const float* fcw = (const float*)d_in[8];
    const float* fcb = (const float*)d_in[9];
    const int*   row = (const int*)d_in[10];
    const int*   col = (const int*)d_in[11];
    float* out = (float*)d_out;

    // Workspace layout (40 MB total):
    //   [0,  8MB)  A_hat : 512 x 64 x 64 f32
    //   [8, 24MB)  h1    : 32768 x 128 f32   (reused for h3: 32768 x 64)
    //   [24,40MB)  h2    : 32768 x 128 f32
    char* ws = (char*)d_ws;
    float* Ahat = (float*)(ws);
    float* h1   = (float*)(ws + (size_t)(8u << 20));
    float* h2   = (float*)(ws + (size_t)(24u << 20));

    build_ahat<<<G_, 256, 0, stream>>>(row, col, ew, Ahat);
    cheb_layer<64, 128, true ><<<G_, 256, 0, stream>>>(x,  Ahat, W1, b1, h1);
    cheb_layer<128, 128, true ><<<G_, 256, 0, stream>>>(h1, Ahat, W2, b2, h2);
    cheb_layer<128, 64,  false><<<G_, 256, 0, stream>>>(h2, Ahat, W3, b3, h1);
    readout<<<G_, 64, 0, stream>>>(h1, fcw, fcb, out);
}